// RbfPerturbedSphereAttractorODE_79104707658136
// MI455X (gfx1250) — compile-verified
//
#include <hip/hip_runtime.h>

typedef __attribute__((ext_vector_type(2))) float v2f;
typedef __attribute__((ext_vector_type(8))) float v8f;

#define MPAD 3392            // 212 * 16 >= padded M (M = 3375 -> 211 tiles = 3376)
#define BLOCK 256            // 8 wave32 waves per block

// phi = exp(-d2 / (2*0.3^2)) = exp2(d2 * -log2(e)/0.18)
#define NEG_LOG2E_OVER_2L2 (-8.0149724f)
#define EPS 1e-10f
#define HUGEF 3.0e38f

__global__ __launch_bounds__(BLOCK)
void rbf_sphere_attractor_kernel(const float* __restrict__ x,
                                 const float* __restrict__ centers,
                                 const float* __restrict__ cu,
                                 const float* __restrict__ cv,
                                 const float* __restrict__ cw,
                                 float* __restrict__ out,
                                 int N, int M)
{
    // Stage per-center WMMA B-operand data in LDS, precomputed:
    //   sB[2c+0] = (-2*c0, -2*c1)   (used by lanes 0..15  : K=0,1)
    //   sB[2c+1] = (-2*c2, |c|^2)   (used by lanes 16..31 : K=2,3)
    __shared__ float2 sB[MPAD * 2];   // 54272 bytes

    const int numTiles = (M + 15) >> 4;
    const int cpad = numTiles << 4;   // padded center count (<= MPAD)

    for (int c = threadIdx.x; c < cpad; c += BLOCK) {
        float c0 = 0.f, c1 = 0.f, c2 = 0.f;
        if (c < M) {
            c0 = centers[3 * c + 0];
            c1 = centers[3 * c + 1];
            c2 = centers[3 * c + 2];
        }
        const float csq = c0 * c0 + c1 * c1 + c2 * c2;
        sB[2 * c + 0] = make_float2(-2.f * c0, -2.f * c1);
        sB[2 * c + 1] = make_float2(-2.f * c2, csq);
    }
    __syncthreads();

    // One wave handles 16 consecutive points.
    const int lane  = threadIdx.x & 31;
    const int wave  = threadIdx.x >> 5;
    const int row0  = (blockIdx.x * (BLOCK >> 5) + wave) << 4;
    if (row0 >= N) return;                 // wave-uniform: EXEC stays all-ones

    const int h   = lane >> 4;             // half-wave: selects K pair / row block
    const int col = lane & 15;             // tile column (center within K-tile)

    // A operand (16x4 f32, loop-invariant): row p = col, K = vgpr + 2*h
    //   V0: h=0 -> x0 ; h=1 -> x2      V1: h=0 -> x1 ; h=1 -> 1.0
    float ax = 0.f, ay = 0.f, az = 0.f;
    {
        const int rA = row0 + col;
        if (rA < N) { ax = x[3*rA]; ay = x[3*rA+1]; az = x[3*rA+2]; }
    }
    v2f A;
    A[0] = h ? az : ax;
    A[1] = h ? 1.0f : ay;

    // |x|^2 for the 8 rows this lane owns in the C/D tile (row = i + 8*h).
    // Seeded as the WMMA C operand so the matrix unit emits dist^2 directly:
    //   D = |x|^2 + |c|^2 - 2 x.c
    float xsq[8];
    v8f Cacc;
#pragma unroll
    for (int i = 0; i < 8; ++i) {
        const int r = row0 + (h << 3) + i;
        float a = 0.f, b = 0.f, c = 0.f;
        if (r < N) { a = x[3*r]; b = x[3*r+1]; c = x[3*r+2]; }
        xsq[i] = a * a + b * b + c * c;
        Cacc[i] = xsq[i];
    }

    float au[8], av[8], aw[8];
#pragma unroll
    for (int i = 0; i < 8; ++i) { au[i] = 0.f; av[i] = 0.f; aw[i] = 0.f; }

    for (int t = 0; t < numTiles; ++t) {
        const int c = (t << 4) + col;

        // B operand (4x16 f32): col n = lane%16, K = vgpr + 2*h  -> one ds_load_b64
        const float2 bb = sB[2 * c + h];
        v2f B;
        B[0] = bb.x;
        B[1] = bb.y;

        // Coefficients for this lane's center column (zero-padded tail)
        float qu = 0.f, qv = 0.f, qw = 0.f;
        if (c < M) { qu = cu[c]; qv = cv[c]; qw = cw[c]; }

        // D[p][n] = |x_p|^2 + |c_n|^2 - 2 x_p.c_n  (full fp32 Gram tile, C = |x|^2)
        v8f D = __builtin_amdgcn_wmma_f32_16x16x4_f32(
            /*neg_a=*/false, A, /*neg_b=*/false, B,
            /*c_mod=*/(short)0, Cacc, /*reuse_a=*/false, /*reuse_b=*/false);

        // max(d2, eps) as a single v_med3_f32 (no canonicalize pass),
        // phi = exp2(d2 * -log2(e)/(2 l^2)); accumulate phi * coeff per row.
#pragma unroll
        for (int i = 0; i < 8; ++i) {
            const float d2  = __builtin_amdgcn_fmed3f(D[i], EPS, HUGEF);
            const float phi = __builtin_amdgcn_exp2f(d2 * NEG_LOG2E_OVER_2L2);
            au[i] = fmaf(phi, qu, au[i]);
            av[i] = fmaf(phi, qv, av[i]);
            aw[i] = fmaf(phi, qw, aw[i]);
        }
    }

    // Reduce the center dimension: butterfly across the 16 lanes of each half.
#pragma unroll
    for (int i = 0; i < 8; ++i) {
#pragma unroll
        for (int m = 1; m < 16; m <<= 1) {
            au[i] += __shfl_xor(au[i], m, 16);
            av[i] += __shfl_xor(av[i], m, 16);
            aw[i] += __shfl_xor(aw[i], m, 16);
        }
    }

    // Lanes 0 and 16 write the 8 rows of their half: out = x*(1-||x||) + uvw
    if (col == 0) {
#pragma unroll
        for (int i = 0; i < 8; ++i) {
            const int r = row0 + (h << 3) + i;
            if (r < N) {
                const float x0 = x[3*r], x1 = x[3*r+1], x2 = x[3*r+2];
                const float s  = 1.0f - sqrtf(xsq[i]);
                out[3*r+0] = fmaf(x0, s, au[i]);
                out[3*r+1] = fmaf(x1, s, av[i]);
                out[3*r+2] = fmaf(x2, s, aw[i]);
            }
        }
    }
}

extern "C" void kernel_launch(void* const* d_in, const int* in_sizes, int n_in,
                              void* d_out, int out_size, void* d_ws, size_t ws_size,
                              hipStream_t stream) {
    (void)n_in; (void)out_size; (void)d_ws; (void)ws_size;
    const float* x       = (const float*)d_in[0];
    const float* centers = (const float*)d_in[1];
    const float* cu      = (const float*)d_in[2];
    const float* cv      = (const float*)d_in[3];
    const float* cw      = (const float*)d_in[4];
    float* out           = (float*)d_out;

    const int N = in_sizes[0] / 3;
    const int M = in_sizes[1] / 3;

    const int wavesPerBlock = BLOCK / 32;
    const int numWaves = (N + 15) / 16;
    const int grid = (numWaves + wavesPerBlock - 1) / wavesPerBlock;

    rbf_sphere_attractor_kernel<<<grid, BLOCK, 0, stream>>>(
        x, centers, cu, cv, cw, out, N, M);
}